// CompetitiveLayer_2_30597347016791
// MI455X (gfx1250) — compile-verified
//
#include <hip/hip_runtime.h>
#include <hip/hip_bf16.h>

typedef __attribute__((ext_vector_type(2))) float v2f;
typedef __attribute__((ext_vector_type(4))) float v4f;
typedef __attribute__((ext_vector_type(8))) float v8f;

#define BTOT   1024
#define NA     256
#define NB     256
#define TM     16          // batch rows per workgroup (WMMA M)
#define NWAVES 16          // waves per workgroup; wave w owns cols [16w,16w+16)
#define NTHREADS (NWAVES * 32)
#define KSTRIDE 260        // LDS row stride in floats (260 % 64 == 4 -> conflict-free)
#define FSTRIDE 260
#define NITERS 51          // 50 no-grad iterations + 1 differentiable iteration

// Fast reciprocal: v_rcp_f32 + one Newton-Raphson step (~0.5 ulp)
__device__ __forceinline__ float fast_rcp(float d) {
    float r = __builtin_amdgcn_rcpf(d);
    float e = __builtin_fmaf(-d, r, 1.0f);
    return __builtin_fmaf(r, e, r);
}

// ---------------------------------------------------------------------------
// Kernel 1: per-16-row fixed-point solver, fully on-chip.
//   LDS: K[256][260] + AF[16][260] + BF[16][260]  = 299,520 B  (< 320 KB WGP LDS)
//   All matmuls via V_WMMA_F32_16X16X4_F32; accumulator seeded with 1.0 so the
//   "+1" of the denominator is free.
// ---------------------------------------------------------------------------
__global__ __launch_bounds__(NTHREADS, 1)
void competitive_solver(const float* __restrict__ AT,
                        const float* __restrict__ BT,
                        const float* __restrict__ K,
                        float* __restrict__ AFout,
                        float* __restrict__ BFout)
{
    extern __shared__ float smem[];
    float* sK  = smem;                       // NA * KSTRIDE
    float* sAF = sK  + NA * KSTRIDE;         // TM * FSTRIDE
    float* sBF = sAF + TM * FSTRIDE;         // TM * FSTRIDE

    const int tid  = threadIdx.x;
    const int lane = tid & 31;
    const int w    = tid >> 5;               // wave id 0..15
    const int bm0  = blockIdx.x * TM;        // first batch row of this tile

    const int half = lane >> 4;              // 0 | 1 (lane group)
    const int l16  = lane & 15;
    const int ncol = 16 * w + l16;           // this lane's output column (n)

    // ---- Stage K into LDS (row stride KSTRIDE), float4 transfers ----
    #pragma unroll
    for (int it = 0; it < (NA * NB / 4) / NTHREADS; ++it) {
        int g   = tid + NTHREADS * it;       // float4 index 0..16383
        int row = g >> 6;                    // 64 float4 per row
        int c4  = (g & 63) << 2;
        v4f kv = *(const v4f*)(K + row * NB + c4);
        *(v4f*)(sK + row * KSTRIDE + c4) = kv;
    }
    // ---- Init BF = BT tile in LDS ----
    #pragma unroll
    for (int it = 0; it < (TM * NB) / NTHREADS; ++it) {
        int g   = tid + NTHREADS * it;       // float index 0..4095
        int row = g >> 8;
        int c   = g & 255;
        sBF[row * FSTRIDE + c] = BT[(bm0 + row) * NB + c];
    }
    __syncthreads();

    // ---- Preload AT/BT into registers in the C/D accumulator layout ----
    // C/D layout: lane, vgpr v  ->  m = v + 8*half, n = ncol
    float at[8], bt[8];
    #pragma unroll
    for (int v = 0; v < 8; ++v) {
        int m = v + 8 * half;
        at[v] = AT[(size_t)(bm0 + m) * NA + ncol];
        bt[v] = BT[(size_t)(bm0 + m) * NB + ncol];
    }

    float afr[8], bfr[8];

    for (int iter = 0; iter < NITERS; ++iter) {
        // ============ Phase A:  denomA = 1 + BF @ K^T  (cols i = ncol) ====
        // A[m][k] = BF[m][kk+k] ; B[k][n] = K[16w+n][kk+k]
        v8f acc = {1.f, 1.f, 1.f, 1.f, 1.f, 1.f, 1.f, 1.f};
        #pragma unroll 8
        for (int kk = 0; kk < NB; kk += 4) {
            v2f a = *(const v2f*)(sBF + l16 * FSTRIDE + kk + 2 * half);
            v2f b = *(const v2f*)(sK + (16 * w + l16) * KSTRIDE + kk + 2 * half);
            acc = __builtin_amdgcn_wmma_f32_16x16x4_f32(
                      false, a, false, b, (short)0, acc, false, false);
        }
        #pragma unroll
        for (int v = 0; v < 8; ++v) afr[v] = at[v] * fast_rcp(acc[v]);
        // publish AF tile (disjoint per wave; phase A only read sBF/sK)
        #pragma unroll
        for (int v = 0; v < 8; ++v)
            sAF[(v + 8 * half) * FSTRIDE + ncol] = afr[v];
        __syncthreads();

        // ============ Phase B:  denomB = 1 + AF @ K   (cols j = ncol) =====
        // A[m][k] = AF[m][kk+k] ; B[k][n] = K[kk+k][16w+n]
        v8f acc2 = {1.f, 1.f, 1.f, 1.f, 1.f, 1.f, 1.f, 1.f};
        #pragma unroll 8
        for (int kk = 0; kk < NA; kk += 4) {
            v2f a = *(const v2f*)(sAF + l16 * FSTRIDE + kk + 2 * half);
            v2f b;
            b.x = sK[(kk + 2 * half + 0) * KSTRIDE + ncol];
            b.y = sK[(kk + 2 * half + 1) * KSTRIDE + ncol];
            acc2 = __builtin_amdgcn_wmma_f32_16x16x4_f32(
                       false, a, false, b, (short)0, acc2, false, false);
        }
        #pragma unroll
        for (int v = 0; v < 8; ++v) bfr[v] = bt[v] * fast_rcp(acc2[v]);
        #pragma unroll
        for (int v = 0; v < 8; ++v)
            sBF[(v + 8 * half) * FSTRIDE + ncol] = bfr[v];
        __syncthreads();
    }

    // ---- Write final AF/BF straight from registers to workspace ----
    #pragma unroll
    for (int v = 0; v < 8; ++v) {
        int m = v + 8 * half;
        AFout[(size_t)(bm0 + m) * NA + ncol] = afr[v];
        BFout[(size_t)(bm0 + m) * NB + ncol] = bfr[v];
    }
}

// ---------------------------------------------------------------------------
// Kernel 2: C[b,i,j] = K[i,j] * AF[b,i] * BF[b,j]  — pure streaming store.
// One float4 along j per thread; K & AF/BF stay L2-resident; C (268 MB) is
// written with non-temporal hint so it doesn't thrash the 192 MB L2.
// ---------------------------------------------------------------------------
__global__ __launch_bounds__(256)
void competitive_outer(const float* __restrict__ K,
                       const float* __restrict__ AF,
                       const float* __restrict__ BF,
                       float* __restrict__ C)
{
    int idx = blockIdx.x * 256 + threadIdx.x;   // float4 index, < 2^24
    int j4  = (idx & 63) << 2;                  // j start (64 float4 per row)
    int i   = (idx >> 6) & 255;
    int b   = idx >> 14;

    v4f kv = *(const v4f*)(K  + i * NB + j4);
    v4f bv = *(const v4f*)(BF + b * NB + j4);
    float a = AF[b * NA + i];
    v4f cv = kv * bv * a;
    __builtin_nontemporal_store(cv, (v4f*)(C + (((size_t)b * NA + i) * NB) + j4));
}

extern "C" void kernel_launch(void* const* d_in, const int* in_sizes, int n_in,
                              void* d_out, int out_size, void* d_ws, size_t ws_size,
                              hipStream_t stream) {
    const float* AT = (const float*)d_in[0];   // [1024, 256]
    const float* BT = (const float*)d_in[1];   // [1024, 256]
    const float* K  = (const float*)d_in[2];   // [256, 256]
    float* out = (float*)d_out;                // [1024, 256, 256]

    float* AFw = (float*)d_ws;                 // 1024*256 floats
    float* BFw = AFw + (size_t)BTOT * NA;      // 1024*256 floats

    size_t lds_bytes = (size_t)(NA * KSTRIDE + 2 * TM * FSTRIDE) * sizeof(float);

    competitive_solver<<<BTOT / TM, NTHREADS, lds_bytes, stream>>>(AT, BT, K, AFw, BFw);

    int n_f4 = (BTOT * NA * NB) / 4;           // 16,777,216
    competitive_outer<<<n_f4 / 256, 256, 0, stream>>>(K, AFw, BFw, out);
}